// InsulationLoss_75488345195120
// MI455X (gfx1250) — compile-verified
//
#include <hip/hip_runtime.h>
#include <hip/hip_bf16.h>

// ---------------------------------------------------------------------------
// InsulationLoss for MI455X (gfx1250, wave32).
// Pipeline (3 kernels, deterministic, no atomics):
//   K1: per image (128 blocks x 512 thr): sliding 51-wide row sums of the
//       |r-c|<=50 band stored in DIAGONAL layout Dg[i][d] = R[i+d][i] in LDS;
//       W[i] = sum_d Dg[i][d] via V_WMMA_F32_16X16X4_F32 (B = ones), with
//       zero-padded LDS so all A-gathers are unconditional ds_load_b64.
//       Tree-reduce block sum -> blockSum[img].
//   K2: per batch b (64 blocks): tensor totals re-summed serially from
//       blockSum, iv = log2(W * B*L / T), 10-wide pools,
//       (dv_tar - dv_out)^2 tree-reduced -> partial[b].
//   K3: 1 block: sum 64 partials, scale by 1/(64*443) -> d_out[0].
// ---------------------------------------------------------------------------

#define NN      512
#define K2W     51
#define LWIN    462          // NN - K2W + 1
#define K1W     10
#define LDV     443          // LWIN - 2*K1W + 1
#define DROWS   464          // padded window rows (29 tiles * 16)
#define RSTRIDE 54           // padded d-stride: even (b64 align), gcd(54,64)=2
#define WSTRIDE 464          // padded per-image W stride in workspace

typedef float v2f __attribute__((ext_vector_type(2)));
typedef float v8f __attribute__((ext_vector_type(8)));

__global__ __launch_bounds__(512)
void band_window_sums(const float* __restrict__ outp,
                      const float* __restrict__ targ,
                      float* __restrict__ Wws,
                      float* __restrict__ blockSum)
{
    __shared__ float Dglds[DROWS * RSTRIDE];   // ~98 KB, diagonal layout
    __shared__ float Wlds[WSTRIDE];
    __shared__ float red[512];

    const int img = blockIdx.x;                        // 0..127
    const float* __restrict__ x =
        (img < 64 ? outp : targ) + (size_t)(img & 63) * NN * NN;

    // ---- Phase 0: zero-fill LDS so padded reads return 0 -----------------
    {
        float4* p = (float4*)Dglds;
        const int tot4 = (DROWS * RSTRIDE) / 4;        // 6264
        for (int k = threadIdx.x; k < tot4; k += 512)
            p[k] = make_float4(0.f, 0.f, 0.f, 0.f);
    }
    __syncthreads();

    // ---- Phase 1: sliding 51-wide row sums, diagonal LDS layout ----------
    // Dg[i][d] = sum_{c=i..i+50} x[i+d][c]; thread r covers i = r-d for
    //   d in [max(0, r-461), min(50, r)]
    {
        const int r   = threadIdx.x;                   // 0..511
        const int dlo = (r > LWIN - 1) ? (r - (LWIN - 1)) : 0;
        const int dhi = (r < K2W - 1) ? r : (K2W - 1);
        const float* __restrict__ row = x + (size_t)r * NN;

        int i = r - dlo;                               // largest window index
        __builtin_prefetch(row + i, 0, 0);             // global_prefetch_b8

        float s = 0.0f;
        for (int c = i; c < i + K2W; ++c) s += row[c];
        Dglds[i * RSTRIDE + dlo] = s;
        for (int d = dlo + 1; d <= dhi; ++d) {
            --i;                                       // window slides left
            s += row[i] - row[i + K2W];
            Dglds[i * RSTRIDE + d] = s;
        }
    }
    __syncthreads();

    // ---- Phase 2: W[i] = sum_{d=0..50} Dg[i][d] via f32 WMMA -------------
    // A[m][k] = Dg[i0+m][k0+k] (adjacent words -> one b64 load), B = ones.
    {
        const int wave  = threadIdx.x >> 5;            // 0..15
        const int lane  = threadIdx.x & 31;
        const int m     = lane & 15;
        const int kbase = (lane >> 4) << 1;            // 0 or 2 (A lane layout)

        v2f bones; bones.x = 1.0f; bones.y = 1.0f;     // layout-independent

        for (int tile = wave; tile < 29; tile += 16) {
            const int i0 = tile * 16;
            const float* arow = &Dglds[(i0 + m) * RSTRIDE + kbase];
            v8f acc = {0.f,0.f,0.f,0.f,0.f,0.f,0.f,0.f};

            #pragma unroll
            for (int k0 = 0; k0 < 52; k0 += 4) {       // d = 0..51 (pads = 0)
                v2f am = *(const v2f*)(arow + k0);     // ds_load_b64
                acc = __builtin_amdgcn_wmma_f32_16x16x4_f32(
                    false, am, false, bones, (short)0, acc, false, false);
            }

            // D layout: VGPR v -> M=v (lanes 0-15, N=lane) / M=8+v (16-31)
            if (lane == 0) {
                #pragma unroll
                for (int v = 0; v < 8; ++v) {
                    const int w = i0 + v;
                    if (w < LWIN) { Wws[img * WSTRIDE + w] = acc[v]; Wlds[w] = acc[v]; }
                }
            } else if (lane == 16) {
                #pragma unroll
                for (int v = 0; v < 8; ++v) {
                    const int w = i0 + 8 + v;
                    if (w < LWIN) { Wws[img * WSTRIDE + w] = acc[v]; Wlds[w] = acc[v]; }
                }
            }
        }
    }
    if (threadIdx.x >= LWIN && threadIdx.x < WSTRIDE) Wlds[threadIdx.x] = 0.0f;
    __syncthreads();

    // ---- Phase 3: deterministic block sum of W ---------------------------
    red[threadIdx.x] = (threadIdx.x < WSTRIDE) ? Wlds[threadIdx.x] : 0.0f;
    __syncthreads();
    for (int off = 256; off > 0; off >>= 1) {
        if ((int)threadIdx.x < off) red[threadIdx.x] += red[threadIdx.x + off];
        __syncthreads();
    }
    if (threadIdx.x == 0) blockSum[img] = red[0];
}

__global__ __launch_bounds__(512)
void insulation_loss_partial(const float* __restrict__ Wws,
                             const float* __restrict__ blockSum,
                             float* __restrict__ partial)
{
    __shared__ float ivo[LWIN];
    __shared__ float ivt[LWIN];
    __shared__ float red[512];

    const int b = blockIdx.x;                          // batch 0..63
    const int t = threadIdx.x;

    // Deterministic tensor totals (identical serial sum in every block).
    float To = 0.0f, Tt = 0.0f;
    for (int k = 0; k < 64; ++k) { To += blockSum[k]; Tt += blockSum[64 + k]; }
    const float so = (64.0f * (float)LWIN) / To;       // W*so == iv/mean(iv)
    const float st = (64.0f * (float)LWIN) / Tt;

    if (t < LWIN) {
        ivo[t] = log2f(Wws[b * WSTRIDE + t] * so);
        ivt[t] = log2f(Wws[(64 + b) * WSTRIDE + t] * st);
    }
    __syncthreads();

    float e2 = 0.0f;
    if (t < LDV) {
        float sbo = 0.f, sto = 0.f, sbt = 0.f, stt = 0.f;
        #pragma unroll
        for (int k = 0; k < K1W; ++k) {
            sbo += ivo[t + k];        sto += ivo[t + K1W + k];
            sbt += ivt[t + k];        stt += ivt[t + K1W + k];
        }
        const float dvo = (sto - sbo) * (1.0f / K1W);
        const float dvt = (stt - sbt) * (1.0f / K1W);
        const float e = dvt - dvo;
        e2 = e * e;
    }
    red[t] = e2;
    __syncthreads();
    for (int off = 256; off > 0; off >>= 1) {
        if (t < off) red[t] += red[t + off];
        __syncthreads();
    }
    if (t == 0) partial[b] = red[0];
}

__global__ __launch_bounds__(64)
void finalize_loss(const float* __restrict__ partial, float* __restrict__ out)
{
    __shared__ float red[64];
    const int t = threadIdx.x;
    red[t] = partial[t];
    __syncthreads();
    for (int off = 32; off > 0; off >>= 1) {
        if (t < off) red[t] += red[t + off];
        __syncthreads();
    }
    if (t == 0) out[0] = red[0] * (1.0f / (64.0f * (float)LDV));
}

extern "C" void kernel_launch(void* const* d_in, const int* in_sizes, int n_in,
                              void* d_out, int out_size, void* d_ws, size_t ws_size,
                              hipStream_t stream)
{
    (void)in_sizes; (void)n_in; (void)out_size; (void)ws_size;
    const float* outp = (const float*)d_in[0];
    const float* targ = (const float*)d_in[1];

    float* ws       = (float*)d_ws;
    float* Wws      = ws;                      // 128 * 464 floats
    float* blockSum = Wws + 128 * WSTRIDE;     // 128 floats
    float* partial  = blockSum + 128;          // 64 floats

    band_window_sums       <<<128, 512, 0, stream>>>(outp, targ, Wws, blockSum);
    insulation_loss_partial<<< 64, 512, 0, stream>>>(Wws, blockSum, partial);
    finalize_loss          <<<  1,  64, 0, stream>>>(partial, (float*)d_out);
}